// NormalizedCrossCorrelation_20675972563795
// MI455X (gfx1250) — compile-verified
//
#include <hip/hip_runtime.h>
#include <hip/hip_bf16.h>

// ---- problem constants (from reference: x,y are [5000, 512, 16] f32) ----
#define COLS   8192          // 512 traces * 16 channels, contiguous innermost
#define VCOLS  2048          // COLS / 4 (float4 lanes)
#define DT_F   0.001f
#define EPS_F  1e-10f
#define CBLOCK 256           // threads per block
#define NCHUNK_MAX 32

typedef __attribute__((ext_vector_type(4))) float    f32x4;
typedef __attribute__((ext_vector_type(4))) unsigned u32x4;
typedef __attribute__((ext_vector_type(2))) float    v2f;
typedef __attribute__((ext_vector_type(8))) float    v8f;

// ---------------------------------------------------------------------------
// Exact f32 wave32 reduction using V_WMMA_F32_16X16X4_F32.
// A[16x4]: a.x = lane value, a.y = 0  (each lane value appears exactly once
// in A, mapping irrelevant).  B[4x16] = all ones (layout-independent).
// D[m][n] = sum_k A[m][k].  Summing a lane's 8 D-VGPRs gives, on lanes 0..15,
// rows 0..7 of the rowsums, and on lanes 16..31, rows 8..15.  Total sum of all
// 32 lane inputs = s(lane0) + s(lane16).  EXEC must be all ones: callers
// invoke this with a fully active wave.
// ---------------------------------------------------------------------------
__device__ __forceinline__ float wave_reduce_wmma(float v) {
    v2f a; a.x = v;   a.y = 0.0f;
    v2f b; b.x = 1.0f; b.y = 1.0f;
    v8f c = {};
    c = __builtin_amdgcn_wmma_f32_16x16x4_f32(
            /*neg_a=*/false, a, /*neg_b=*/false, b,
            /*c_mod=*/(short)0, c, /*reuse_a=*/false, /*reuse_b=*/false);
    float s = ((c[0] + c[1]) + (c[2] + c[3])) + ((c[4] + c[5]) + (c[6] + c[7]));
    return __shfl(s, 0, 32) + __shfl(s, 16, 32);
}

// ---------------------------------------------------------------------------
// Kernel 1: streaming partial reduction over an n-chunk.
// grid = (8 column-blocks, nchunk), block = 256.  Each lane owns 4 columns
// (one float4) and marches down the time axis with non-temporal b128 loads.
// ---------------------------------------------------------------------------
__global__ __launch_bounds__(CBLOCK)
void ncc_partial(const f32x4* __restrict__ x4, const f32x4* __restrict__ y4,
                 f32x4* __restrict__ wsXX, f32x4* __restrict__ wsYY,
                 f32x4* __restrict__ wsXY, u32x4* __restrict__ wsM,
                 int nt, int steps) {
    const int vcol  = blockIdx.x * CBLOCK + threadIdx.x;   // 0..2047
    const int chunk = blockIdx.y;
    int n0 = chunk * steps;
    int n1 = n0 + steps; if (n1 > nt) n1 = nt;

    f32x4 sxx = {0.f,0.f,0.f,0.f}, syy = {0.f,0.f,0.f,0.f}, sxy = {0.f,0.f,0.f,0.f};
    u32x4 mb  = {0u,0u,0u,0u};

    const f32x4* xp = x4 + (size_t)n0 * VCOLS + vcol;
    const f32x4* yp = y4 + (size_t)n0 * VCOLS + vcol;

    #pragma unroll 4
    for (int n = n0; n < n1; ++n) {
        f32x4 xv = __builtin_nontemporal_load(xp);   // streaming: no reuse
        f32x4 yv = __builtin_nontemporal_load(yp);
        __builtin_prefetch(xp + 32 * VCOLS, 0, 0);   // global_prefetch_b8
        __builtin_prefetch(yp + 32 * VCOLS, 0, 0);

        sxx.x = fmaf(xv.x, xv.x, sxx.x); sxx.y = fmaf(xv.y, xv.y, sxx.y);
        sxx.z = fmaf(xv.z, xv.z, sxx.z); sxx.w = fmaf(xv.w, xv.w, sxx.w);
        syy.x = fmaf(yv.x, yv.x, syy.x); syy.y = fmaf(yv.y, yv.y, syy.y);
        syy.z = fmaf(yv.z, yv.z, syy.z); syy.w = fmaf(yv.w, yv.w, syy.w);
        sxy.x = fmaf(xv.x, yv.x, sxy.x); sxy.y = fmaf(xv.y, yv.y, sxy.y);
        sxy.z = fmaf(xv.z, yv.z, sxy.z); sxy.w = fmaf(xv.w, yv.w, sxy.w);

        // max|x| > 0  <=>  OR of (bits & ~sign) != 0
        mb.x |= __float_as_uint(xv.x) & 0x7fffffffu;
        mb.y |= __float_as_uint(xv.y) & 0x7fffffffu;
        mb.z |= __float_as_uint(xv.z) & 0x7fffffffu;
        mb.w |= __float_as_uint(xv.w) & 0x7fffffffu;

        xp += VCOLS; yp += VCOLS;
    }

    const size_t o = (size_t)chunk * VCOLS + vcol;   // b128 stores
    wsXX[o] = sxx; wsYY[o] = syy; wsXY[o] = sxy; wsM[o] = mb;
}

// ---------------------------------------------------------------------------
// Kernel 2: per-column combine across n-chunks, compute cc, block-reduce.
// grid = 32, block = 256 (one thread per column).  Per-wave reduction via
// WMMA, cross-wave via 8-entry LDS.
// ---------------------------------------------------------------------------
__global__ __launch_bounds__(CBLOCK)
void ncc_combine(const float* __restrict__ XX, const float* __restrict__ YY,
                 const float* __restrict__ XY, const unsigned* __restrict__ M,
                 float* __restrict__ blockres, int nchunk) {
    const int col = blockIdx.x * CBLOCK + threadIdx.x;    // 0..8191
    float sxx = 0.f, syy = 0.f, sxy = 0.f; unsigned mb = 0u;
    for (int c = 0; c < nchunk; ++c) {
        const size_t o = (size_t)c * COLS + col;
        sxx += XX[o]; syy += YY[o]; sxy += XY[o]; mb |= M[o];
    }
    const float xn = sqrtf(sxx) + EPS_F;
    const float yn = sqrtf(syy) + EPS_F;
    const float cc = (mb != 0u) ? (-DT_F * sxy / (xn * yn)) : 0.0f;

    const float t = wave_reduce_wmma(cc);                 // full-EXEC wave

    __shared__ float lds[8];
    const int lane = threadIdx.x & 31, wid = threadIdx.x >> 5;
    if (lane == 0) lds[wid] = t;
    __syncthreads();
    if (threadIdx.x == 0) {
        float b = 0.f;
        #pragma unroll
        for (int i = 0; i < 8; ++i) b += lds[i];
        blockres[blockIdx.x] = b;
    }
}

// ---------------------------------------------------------------------------
// Kernel 3: final 32 -> 1 with a single fully-active wave + WMMA reduce.
// ---------------------------------------------------------------------------
__global__ __launch_bounds__(32)
void ncc_final(const float* __restrict__ blockres, float* __restrict__ out) {
    const float v = blockres[threadIdx.x];
    const float t = wave_reduce_wmma(v);
    if (threadIdx.x == 0) out[0] = t;
}

// ---------------------------------------------------------------------------
extern "C" void kernel_launch(void* const* d_in, const int* in_sizes, int n_in,
                              void* d_out, int out_size, void* d_ws, size_t ws_size,
                              hipStream_t stream) {
    const f32x4* x4 = (const f32x4*)d_in[0];
    const f32x4* y4 = (const f32x4*)d_in[1];
    float* out = (float*)d_out;

    const int nt = in_sizes[0] / COLS;                    // 5000

    // Workspace layout (floats): XX | YY | XY | M(u32) | blockres(32)
    // per chunk: 4 arrays * 8192 * 4B = 128 KiB.  Adapt chunk count to ws_size
    // (deterministic: depends only on ws_size).
    const size_t per_chunk_bytes = (size_t)4 * COLS * sizeof(float);
    int nchunk = NCHUNK_MAX;
    if (ws_size < (size_t)nchunk * per_chunk_bytes + 256) {
        size_t avail = ws_size > 256 ? ws_size - 256 : 0;
        nchunk = (int)(avail / per_chunk_bytes);
        if (nchunk < 1)  nchunk = 1;
        if (nchunk > NCHUNK_MAX) nchunk = NCHUNK_MAX;
    }
    const int steps = (nt + nchunk - 1) / nchunk;

    float* f = (float*)d_ws;
    f32x4* wsXX = (f32x4*)(f);
    f32x4* wsYY = (f32x4*)(f + (size_t)1 * nchunk * COLS);
    f32x4* wsXY = (f32x4*)(f + (size_t)2 * nchunk * COLS);
    u32x4* wsM  = (u32x4*)(f + (size_t)3 * nchunk * COLS);
    float* blockres =      f + (size_t)4 * nchunk * COLS;

    dim3 g1(VCOLS / CBLOCK, nchunk);                      // 8 x nchunk blocks
    ncc_partial<<<g1, CBLOCK, 0, stream>>>(x4, y4, wsXX, wsYY, wsXY, wsM, nt, steps);

    ncc_combine<<<COLS / CBLOCK, CBLOCK, 0, stream>>>(
        (const float*)wsXX, (const float*)wsYY, (const float*)wsXY,
        (const unsigned*)wsM, blockres, nchunk);

    ncc_final<<<1, 32, 0, stream>>>(blockres, out);
}